// GraphEncoderNetwork_37426345017674
// MI455X (gfx1250) — compile-verified
//
#include <hip/hip_runtime.h>

#define N_NODES 200000
#define N_EDGES 6400000
#define N_DAGS  2000
#define N_JOBS  50

typedef __attribute__((ext_vector_type(2))) float v2f;
typedef __attribute__((ext_vector_type(8))) float v8f;

// ---------------------------------------------------------------------------
// One MLP layer on a 16-row tile held in per-wave LDS.
//   Y[16 x NOUT] = act(X[16 x DIN] @ W[DIN x NOUT] + b)
// Uses chained V_WMMA_F32_16X16X4_F32 (full fp32 precision).
// A-frag (16x4 f32): lane(lo,hi), vgpr v -> X[lo][4*kt + 2*hi + v]
// B-frag (4x16 f32): lane(lo,hi), vgpr v -> W[4*kt + 2*hi + v][n]
// C/D  (16x16 f32): lane(lo,hi), vgpr v -> Y[v + 8*hi][lo]
// Result is written back into the same LDS tile (same-wave DS ops are
// in-order on CDNA5, so no barrier is required).
// ---------------------------------------------------------------------------
template <int DIN, int NOUT, int TS, bool RELU>
__device__ __forceinline__ void wave_layer(float* tile,
                                           const float* __restrict__ W,
                                           const float* __restrict__ B) {
  const int lane = threadIdx.x & 31;
  const int lo = lane & 15;
  const int hi = lane >> 4;
  constexpr int KT = DIN / 4;
  constexpr int NT = (NOUT + 15) / 16;

  v2f a[KT];
#pragma unroll
  for (int kt = 0; kt < KT; ++kt) {
    const int k = kt * 4 + 2 * hi;
    a[kt].x = tile[lo * TS + k];
    a[kt].y = tile[lo * TS + k + 1];
  }

  v8f acc[NT];
#pragma unroll
  for (int nt = 0; nt < NT; ++nt) {
    const int n = nt * 16 + lo;
    v8f c = {};
#pragma unroll
    for (int kt = 0; kt < KT; ++kt) {
      const int k = kt * 4 + 2 * hi;
      v2f b;
      b.x = (n < NOUT) ? W[(k + 0) * NOUT + n] : 0.0f;
      b.y = (n < NOUT) ? W[(k + 1) * NOUT + n] : 0.0f;
      c = __builtin_amdgcn_wmma_f32_16x16x4_f32(false, a[kt], false, b,
                                                (short)0, c, false, false);
    }
    acc[nt] = c;
  }

#pragma unroll
  for (int nt = 0; nt < NT; ++nt) {
    const int n = nt * 16 + lo;
    if (n < NOUT) {
      const float bias = B[n];
#pragma unroll
      for (int v = 0; v < 8; ++v) {
        float y = acc[nt][v] + bias;
        if (RELU) y = fmaxf(y, 0.0f);
        tile[(v + 8 * hi) * TS + n] = y;
      }
    }
  }
}

// Full 3-layer MLP (DIN -> 32 -> 16 -> DOUT), one 16-row tile per wave.
template <int DIN, int DOUT, int TS>
__global__ void __launch_bounds__(256)
k_mlp(const float* __restrict__ in, float* __restrict__ out, int nrows,
      const float* __restrict__ W1, const float* __restrict__ b1,
      const float* __restrict__ W2, const float* __restrict__ b2,
      const float* __restrict__ W3, const float* __restrict__ b3) {
  __shared__ float smem[8 * 16 * TS];
  const int wave = threadIdx.x >> 5;
  const int lane = threadIdx.x & 31;
  const int tile_idx = blockIdx.x * 8 + wave;
  const int ntiles = (nrows + 15) >> 4;
  if (tile_idx >= ntiles) return;  // wave-uniform: EXEC stays all-ones for WMMA

  float* tile = smem + wave * 16 * TS;
  const int base = tile_idx * 16;

  // Stage input tile (rows past nrows zero-padded).
  for (int idx = lane; idx < 16 * DIN; idx += 32) {
    const int r = idx / DIN, c = idx % DIN;
    tile[r * TS + c] = (base + r < nrows) ? in[(size_t)(base + r) * DIN + c]
                                          : 0.0f;
  }

  wave_layer<DIN, 32, TS, true>(tile, W1, b1);
  wave_layer<32, 16, TS, true>(tile, W2, b2);
  wave_layer<16, DOUT, TS, false>(tile, W3, b3);

  for (int idx = lane; idx < 16 * DOUT; idx += 32) {
    const int r = idx / DOUT, c = idx % DOUT;
    if (base + r < nrows) out[(size_t)(base + r) * DOUT + c] = tile[r * TS + c];
  }
}

// ---------------------------------------------------------------------------
// Graph kernels
// ---------------------------------------------------------------------------
__global__ void k_init_deg(float* __restrict__ deg) {
  int i = blockIdx.x * blockDim.x + threadIdx.x;
  if (i < N_NODES) deg[i] = 1.0f;  // self-loop contribution
}

__global__ void k_count_deg(const int* __restrict__ col, float* __restrict__ deg) {
  int e = blockIdx.x * blockDim.x + threadIdx.x;
  if (e < N_EDGES) atomicAdd(&deg[col[e]], 1.0f);
}

__global__ void k_dinv(float* __restrict__ deg) {
  int i = blockIdx.x * blockDim.x + threadIdx.x;
  if (i < N_NODES) {
    float d = deg[i];
    deg[i] = (d > 0.0f) ? rsqrtf(d) : 0.0f;  // deg >= 1 always (self loops)
  }
}

// agg[i][f] = dinv[i]^2 * h[i][f]  (self-loop term doubles as initialization)
__global__ void k_agg_init(const float* __restrict__ h,
                           const float* __restrict__ dinv,
                           float* __restrict__ agg) {
  int t = blockIdx.x * blockDim.x + threadIdx.x;
  if (t < N_NODES * 8) {
    int i = t >> 3;
    float d = dinv[i];
    agg[t] = d * d * h[t];
  }
}

// agg[row][f] += dinv[row]*dinv[col]*h[col][f] ; one thread per (edge, feature)
__global__ void k_edge_scatter(const int* __restrict__ row,
                               const int* __restrict__ col,
                               const float* __restrict__ h,
                               const float* __restrict__ dinv,
                               float* __restrict__ agg) {
  long long t = (long long)blockIdx.x * blockDim.x + threadIdx.x;
  if (t < (long long)N_EDGES * 8) {
    int e = (int)(t >> 3);
    int f = (int)(t & 7);
    int r = row[e], c = col[e];
    float w = dinv[r] * dinv[c];
    atomicAdd(&agg[(size_t)r * 8 + f], w * h[(size_t)c * 8 + f]);
  }
}

__global__ void k_zero(float* __restrict__ p, int n) {
  int i = blockIdx.x * blockDim.x + threadIdx.x;
  if (i < n) p[i] = 0.0f;
}

// ypre[batch[i]][c] += concat(x, x_node)[i][c]; one thread per (node, channel)
__global__ void k_pool_nodes(const float* __restrict__ x,
                             const float* __restrict__ xn,
                             const int* __restrict__ batch,
                             float* __restrict__ ypre) {
  int t = blockIdx.x * blockDim.x + threadIdx.x;
  if (t < N_NODES * 64) {
    int i = t >> 6;
    int c = t & 63;
    float v = (c < 32) ? x[(size_t)i * 32 + c] : xn[(size_t)i * 32 + (c - 32)];
    atomicAdd(&ypre[(size_t)batch[i] * 64 + c], v);
  }
}

// zpre[job(d)][c] += y[d][c]; job via upper_bound(indptr, d) - 1
__global__ void k_pool_dags(const float* __restrict__ y,
                            const int* __restrict__ indptr,
                            float* __restrict__ zpre) {
  int t = blockIdx.x * blockDim.x + threadIdx.x;
  if (t < N_DAGS * 32) {
    int d = t >> 5;
    int c = t & 31;
    int l = 0, r = N_JOBS + 1;
    while (l < r) {
      int m = (l + r) >> 1;
      if (indptr[m] <= d) l = m + 1; else r = m;
    }
    int j = l - 1;
    atomicAdd(&zpre[(size_t)j * 32 + c], y[(size_t)d * 32 + c]);
  }
}

// ---------------------------------------------------------------------------
// Launch
// ---------------------------------------------------------------------------
extern "C" void kernel_launch(void* const* d_in, const int* in_sizes, int n_in,
                              void* d_out, int out_size, void* d_ws,
                              size_t ws_size, hipStream_t stream) {
  (void)in_sizes; (void)n_in; (void)out_size; (void)ws_size;

  const float* x     = (const float*)d_in[0];
  const int*   eidx  = (const int*)d_in[1];
  const int*   batch = (const int*)d_in[2];
  const int*   iptr  = (const int*)d_in[3];
  auto W = [&](int i) { return (const float*)d_in[i]; };
  // d_in[4..9]=p_conv1, [10..15]=p_conv2, [16..21]=p_dag, [22..27]=p_glob

  float* out   = (float*)d_out;
  float* xnode = out;                                // [N_NODES, 32]
  float* yout  = out + (size_t)N_NODES * 32;         // [N_DAGS, 32]
  float* zout  = yout + (size_t)N_DAGS * 32;         // [N_JOBS, 32]

  float* ws   = (float*)d_ws;
  float* deg  = ws;                                  // N_NODES (becomes dinv)
  float* h    = deg + N_NODES;                       // N_NODES*8
  float* agg  = h + (size_t)N_NODES * 8;             // N_NODES*8
  float* ypre = agg + (size_t)N_NODES * 8;           // N_DAGS*64
  float* zpre = ypre + (size_t)N_DAGS * 64;          // N_JOBS*32

  const int* row = eidx;
  const int* col = eidx + N_EDGES;

  const int B = 256;
  const int gb_nodes = (N_NODES + B - 1) / B;
  const int gb_edges = (N_EDGES + B - 1) / B;
  const int gb_mlp_n = (((N_NODES + 15) / 16) + 7) / 8;

  // 1) degrees + dinv
  k_init_deg<<<gb_nodes, B, 0, stream>>>(deg);
  k_count_deg<<<gb_edges, B, 0, stream>>>(col, deg);
  k_dinv<<<gb_nodes, B, 0, stream>>>(deg);

  // 2) h = mlp1(x) : [N, 32] -> [N, 8]
  k_mlp<32, 8, 36><<<gb_mlp_n, B, 0, stream>>>(x, h, N_NODES,
      W(4), W(5), W(6), W(7), W(8), W(9));

  // 3) agg = self-loop init + edge scatter
  k_agg_init<<<(N_NODES * 8 + B - 1) / B, B, 0, stream>>>(h, deg, agg);
  k_edge_scatter<<<(int)(((long long)N_EDGES * 8 + B - 1) / B), B, 0, stream>>>(
      row, col, h, deg, agg);

  // 4) x_node = mlp2(agg) : [N, 8] -> [N, 32]
  k_mlp<8, 32, 36><<<gb_mlp_n, B, 0, stream>>>(agg, xnode, N_NODES,
      W(10), W(11), W(12), W(13), W(14), W(15));

  // 5) ypre = segment_sum(concat(x, x_node), batch) : [N_DAGS, 64]
  k_zero<<<(N_DAGS * 64 + B - 1) / B, B, 0, stream>>>(ypre, N_DAGS * 64);
  k_pool_nodes<<<(N_NODES * 64 + B - 1) / B, B, 0, stream>>>(x, xnode, batch, ypre);

  // 6) y = mlp_dag(ypre) : [N_DAGS, 64] -> [N_DAGS, 32]
  k_mlp<64, 32, 68><<<(((N_DAGS + 15) / 16) + 7) / 8, B, 0, stream>>>(
      ypre, yout, N_DAGS, W(16), W(17), W(18), W(19), W(20), W(21));

  // 7) zpre = segment_sum(y, job_indptr) : [N_JOBS, 32]
  k_zero<<<(N_JOBS * 32 + B - 1) / B, B, 0, stream>>>(zpre, N_JOBS * 32);
  k_pool_dags<<<(N_DAGS * 32 + B - 1) / B, B, 0, stream>>>(yout, iptr, zpre);

  // 8) z = mlp_glob(zpre) : [N_JOBS, 32] -> [N_JOBS, 32]
  k_mlp<32, 32, 36><<<1, B, 0, stream>>>(zpre, zout, N_JOBS,
      W(22), W(23), W(24), W(25), W(26), W(27));
}